// LightGCN_71657234366493
// MI455X (gfx1250) — compile-verified
//
#include <hip/hip_runtime.h>
#include <hip/hip_bf16.h>

#define EMB 64
#define NLAYER 3
#define LSTRIDE (EMB * (NLAYER + 1))   // 256 floats per node in "stacked"

typedef __attribute__((ext_vector_type(2))) float v2f;
typedef __attribute__((ext_vector_type(8))) float v8f;

// ---------------------------------------------------------------------------
// Layer 0: copy embeddings into stacked[n, 0, :]
// ---------------------------------------------------------------------------
__global__ void k_copy_layer0(const float* __restrict__ emb,
                              float* __restrict__ stacked, int N) {
    long t = (long)blockIdx.x * blockDim.x + threadIdx.x;   // one float4 each
    long total = (long)N * (EMB / 4);
    if (t >= total) return;
    int n = (int)(t >> 4);      // EMB/4 == 16 float4 per node
    int q = (int)(t & 15);
    const float4* src = (const float4*)(emb + (size_t)n * EMB);
    float4*       dst = (float4*)(stacked + (size_t)n * LSTRIDE);
    dst[q] = src[q];
}

// ---------------------------------------------------------------------------
// CSR build: zero counts -> histogram -> scan -> fill
// ---------------------------------------------------------------------------
__global__ void k_zero_i32(int* __restrict__ p, int n) {
    int t = blockIdx.x * blockDim.x + threadIdx.x;
    if (t < n) p[t] = 0;
}

__global__ void k_hist(const int* __restrict__ rows, int* __restrict__ counts, int E) {
    int e = blockIdx.x * blockDim.x + threadIdx.x;
    if (e < E) atomicAdd(&counts[rows[e]], 1);
}

// single-block exclusive scan over N bins (N ~ 100K -> ~98 chunks of 1024)
__global__ void k_scan(const int* __restrict__ counts,
                       int* __restrict__ row_ptr,
                       int* __restrict__ cursor, int N) {
    __shared__ int sdata[1024];
    __shared__ int carry;
    int tid = threadIdx.x;
    if (tid == 0) carry = 0;
    __syncthreads();
    for (int base = 0; base < N; base += 1024) {
        int i = base + tid;
        int v = (i < N) ? counts[i] : 0;
        sdata[tid] = v;
        __syncthreads();
        for (int off = 1; off < 1024; off <<= 1) {
            int t = (tid >= off) ? sdata[tid - off] : 0;
            __syncthreads();
            sdata[tid] += t;
            __syncthreads();
        }
        int excl = sdata[tid] - v;
        if (i < N) {
            int rp = carry + excl;
            row_ptr[i] = rp;
            cursor[i]  = rp;
        }
        __syncthreads();
        if (tid == 1023) carry += sdata[1023];
        __syncthreads();
    }
    if (tid == 0) row_ptr[N] = carry;
}

__global__ void k_fill(const int* __restrict__ rows, const int* __restrict__ cols,
                       const float* __restrict__ vals, int* __restrict__ cursor,
                       int* __restrict__ csr_col, float* __restrict__ csr_val, int E) {
    int e = blockIdx.x * blockDim.x + threadIdx.x;
    if (e >= E) return;
    int r = rows[e];
    int p = atomicAdd(&cursor[r], 1);
    csr_col[p] = cols[e];
    csr_val[p] = vals[e];
}

// ---------------------------------------------------------------------------
// SpMM, CSR path: one wave32 per destination row, float2 per lane (64 dims).
// No atomics; software-pipelined edge loop with gfx1250 global_prefetch.
// src/dst point at stacked[*, layer, 0] slices (node stride LSTRIDE floats).
// ---------------------------------------------------------------------------
__global__ void k_spmm_csr(const int* __restrict__ row_ptr,
                           const int* __restrict__ csr_col,
                           const float* __restrict__ csr_val,
                           const float* __restrict__ src,
                           float* __restrict__ dst, int N) {
    int wave = (blockIdx.x * blockDim.x + threadIdx.x) >> 5;
    int lane = threadIdx.x & 31;
    if (wave >= N) return;                    // wave-uniform exit
    int s = row_ptr[wave];
    int t = row_ptr[wave + 1];
    float ax = 0.f, ay = 0.f;
    int e = s;
    if (e < t) {
        int   c = csr_col[e];
        float v = csr_val[e];
        for (; e + 1 < t; ++e) {
            int   cn = csr_col[e + 1];
            float vn = csr_val[e + 1];
            __builtin_prefetch(src + (size_t)cn * LSTRIDE + lane * 2, 0, 0);
            float2 x = *(const float2*)(src + (size_t)c * LSTRIDE + lane * 2);
            ax = fmaf(v, x.x, ax);
            ay = fmaf(v, x.y, ay);
            c = cn; v = vn;
        }
        float2 x = *(const float2*)(src + (size_t)c * LSTRIDE + lane * 2);
        ax = fmaf(v, x.x, ax);
        ay = fmaf(v, x.y, ay);
    }
    float2 out; out.x = ax; out.y = ay;
    *(float2*)(dst + (size_t)wave * LSTRIDE + lane * 2) = out;
}

// ---------------------------------------------------------------------------
// SpMM fallback (workspace too small): per-edge gather + fp32 atomics.
// ---------------------------------------------------------------------------
__global__ void k_zero_layers(float* __restrict__ stacked, int N) {
    long t = (long)blockIdx.x * blockDim.x + threadIdx.x;   // float4 granules
    long total = (long)N * (NLAYER * EMB / 4);              // 48 per node
    if (t >= total) return;
    int n = (int)(t / 48);
    int q = (int)(t % 48);
    float4* p = (float4*)(stacked + (size_t)n * LSTRIDE + EMB);
    p[q] = make_float4(0.f, 0.f, 0.f, 0.f);
}

__global__ void k_spmm_atomic(const int* __restrict__ rows, const int* __restrict__ cols,
                              const float* __restrict__ vals,
                              const float* __restrict__ src,
                              float* __restrict__ dst, int E) {
    long t = (long)blockIdx.x * blockDim.x + threadIdx.x;
    int e = (int)(t >> 4);                    // 16 lanes per edge, float4 each
    if (e >= E) return;
    int q = (int)(t & 15);
    int r = rows[e], c = cols[e];
    float v = vals[e];
    float4 x = *(const float4*)(src + (size_t)c * LSTRIDE + q * 4);
    float* d = dst + (size_t)r * LSTRIDE + q * 4;
    atomicAdd(d + 0, v * x.x);
    atomicAdd(d + 1, v * x.y);
    atomicAdd(d + 2, v * x.z);
    atomicAdd(d + 3, v * x.w);
}

// ---------------------------------------------------------------------------
// Mean over the 4 layer embeddings via V_WMMA_F32_16X16X4_F32.
// D = A x B with A = 0.25 * ones(16x4) (constant -> layout-safe) and
// B(4x16): column j holds the 4 layer values of dim d0+j for node n.
// Because A is constant and the K-contraction is symmetric, any K-slot
// permutation in B's register layout yields the same D. Row 0 of D
// (lanes 0..15, VGPR0) = 16 contiguous mean outputs.
// One wave32 per node, 4 WMMAs (four 16-dim chunks).
// ---------------------------------------------------------------------------
__global__ void k_mean_wmma(const float* __restrict__ stacked,
                            float* __restrict__ mean_out, int N) {
    int wave = (blockIdx.x * blockDim.x + threadIdx.x) >> 5;
    int lane = threadIdx.x & 31;
    if (wave >= N) return;                    // wave-uniform
    const float* base = stacked + (size_t)wave * LSTRIDE;
    int j = lane & 15;
    int half = lane >> 4;                     // lanes 0-15: layers {0,1}; 16-31: {2,3}
    v2f a = {0.25f, 0.25f};
#pragma unroll
    for (int d0 = 0; d0 < EMB; d0 += 16) {
        v2f b;
        b.x = base[(half * 2 + 0) * EMB + d0 + j];
        b.y = base[(half * 2 + 1) * EMB + d0 + j];
        v8f c = {};
        v8f d = __builtin_amdgcn_wmma_f32_16x16x4_f32(
            /*neg_a=*/false, a, /*neg_b=*/false, b,
            /*c_mod=*/(short)0, c, /*reuse_a=*/false, /*reuse_b=*/false);
        if (half == 0) mean_out[(size_t)wave * EMB + d0 + j] = d[0];
    }
}

// ---------------------------------------------------------------------------
extern "C" void kernel_launch(void* const* d_in, const int* in_sizes, int n_in,
                              void* d_out, int out_size, void* d_ws, size_t ws_size,
                              hipStream_t stream) {
    const float* emb  = (const float*)d_in[0];
    const int*   rows = (const int*)d_in[1];
    const int*   cols = (const int*)d_in[2];
    const float* vals = (const float*)d_in[3];
    const int N = in_sizes[0] / EMB;
    const int E = in_sizes[1];

    float* mean_out = (float*)d_out;
    float* stacked  = mean_out + (size_t)N * EMB;   // [N, 4, 64]

    // layer 0 copy
    {
        long total = (long)N * (EMB / 4);
        k_copy_layer0<<<(total + 255) / 256, 256, 0, stream>>>(emb, stacked, N);
    }

    size_t need = ((size_t)3 * N + 1) * sizeof(int) + (size_t)E * 8;
    if (ws_size >= need) {
        int*   counts  = (int*)d_ws;
        int*   cursor  = counts + N;
        int*   row_ptr = cursor + N;
        int*   csr_col = row_ptr + (N + 1);
        float* csr_val = (float*)(csr_col + E);

        k_zero_i32<<<(N + 255) / 256, 256, 0, stream>>>(counts, N);
        k_hist<<<(E + 255) / 256, 256, 0, stream>>>(rows, counts, E);
        k_scan<<<1, 1024, 0, stream>>>(counts, row_ptr, cursor, N);
        k_fill<<<(E + 255) / 256, 256, 0, stream>>>(rows, cols, vals, cursor,
                                                    csr_col, csr_val, E);
        int blocks = (N + 7) / 8;   // 8 waves (rows) per 256-thread block
        for (int l = 1; l <= NLAYER; ++l) {
            k_spmm_csr<<<blocks, 256, 0, stream>>>(
                row_ptr, csr_col, csr_val,
                stacked + (size_t)(l - 1) * EMB, stacked + (size_t)l * EMB, N);
        }
    } else {
        long ztot = (long)N * (NLAYER * EMB / 4);
        k_zero_layers<<<(ztot + 255) / 256, 256, 0, stream>>>(stacked, N);
        long ttot = (long)E * 16;
        for (int l = 1; l <= NLAYER; ++l) {
            k_spmm_atomic<<<(ttot + 255) / 256, 256, 0, stream>>>(
                rows, cols, vals,
                stacked + (size_t)(l - 1) * EMB, stacked + (size_t)l * EMB, E);
        }
    }

    k_mean_wmma<<<(N + 7) / 8, 256, 0, stream>>>(stacked, mean_out, N);
}